// Retina_59304908423288
// MI455X (gfx1250) — compile-verified
//
#include <hip/hip_runtime.h>

typedef unsigned int u32;
typedef float __attribute__((ext_vector_type(4))) f32x4;
typedef u32   __attribute__((ext_vector_type(4))) u32x4;
typedef int   __attribute__((ext_vector_type(4))) i32x4;
typedef int   __attribute__((ext_vector_type(8))) i32x8;

#define FULL_ELEMS  (32 * 3 * 1024 * 1024)   // 100,663,296 floats
#define PATCH_ELEMS (32 * 3 * 128 * 128)     //   1,572,864 floats

// ---------------------------------------------------------------------------
// Kernel 1: write all of `full` (384 MiB). full = x masked to the per-batch
// valid window, zeros elsewhere. 4096 floats per block, 4x B128 NT stores per
// thread. Window test per 4-float group; per-element mask only at the edges.
// ---------------------------------------------------------------------------
__global__ __launch_bounds__(256) void k_full(const float* __restrict__ x,
                                              const int* __restrict__ l,
                                              float* __restrict__ full) {
  const int base = blockIdx.x << 12;      // first float index of this block
  const int bc   = base >> 20;            // uniform per block (4096 | 2^20)
  const int b    = bc / 3;
  const int l0   = l[2 * b];
  const int l1   = l[2 * b + 1];
  const int r_lo = l0 - 193;              // window row start
  const int c_lo = l1 - 193;              // window col start
#pragma unroll
  for (int k = 0; k < 4; ++k) {
    const int f  = base + (((k << 8) + (int)threadIdx.x) << 2); // multiple of 4
    const int r  = (f >> 10) & 1023;
    const int c0 = f & 1023;
    const int dr = r - r_lo;
    const int dc = c0 - c_lo;
    f32x4 v = {0.f, 0.f, 0.f, 0.f};
    if ((u32)dr < 128u && dc > -4 && dc < 128) {
      const f32x4 xv = *(const f32x4*)(x + f);
#pragma unroll
      for (int e = 0; e < 4; ++e)
        if ((u32)(dc + e) < 128u) v[e] = xv[e];
    }
    __builtin_nontemporal_store(v, (f32x4*)(full + f));
  }
}

// ---------------------------------------------------------------------------
// Kernel 2: one block per (b,c). TDM loads the clipped nr x nc window of x
// into LDS (2D tile, row stride 1024), then threads emit the 128x128 patch
// with zero padding outside the valid sub-rectangle.
// ---------------------------------------------------------------------------
__global__ __launch_bounds__(256) void k_patch(const float* __restrict__ x,
                                               const int* __restrict__ l,
                                               float* __restrict__ patch) {
  __shared__ float tile[128 * 128];       // 64 KiB
  const int bc = blockIdx.x;              // 0..95
  const int b  = bc / 3;
  const int l0 = l[2 * b];
  const int l1 = l[2 * b + 1];
  const int r_lo = l0 - 193, c_lo = l1 - 193;
  const int r0 = r_lo < 0 ? 0 : r_lo;
  const int c0 = c_lo < 0 ? 0 : c_lo;
  const int r1 = (l0 - 66) > 1023 ? 1023 : (l0 - 66);
  const int c1 = (l1 - 66) > 1023 ? 1023 : (l1 - 66);
  int nr = r1 - r0 + 1;                   // rows actually inside the image
  int nc = c1 - c0 + 1;                   // cols actually inside the image
  if (nr < 0) nr = 0;
  if (nc < 0) nc = 0;
  const int pt = r0 - r_lo;               // top zero-padding in patch coords
  const int pl = c0 - c_lo;               // left zero-padding in patch coords

  if (nr > 0 && nc > 0 && threadIdx.x < 32) {   // wave 0 issues the TDM
    const unsigned long long ga = (unsigned long long)(const void*)
        (x + (((long long)bc << 20) | ((long long)r0 << 10) | (long long)c0));
    const u32 lds_base = (u32)(unsigned long long)(const void*)&tile[0];

    // ---- D# group 0 (128b): count=1 | lds_addr | global_addr[56:0] | type=2
    u32x4 g0;
    g0.x = 0x1u;
    g0.y = lds_base;
    g0.z = (u32)(ga & 0xFFFFFFFFull);
    g0.w = (u32)((ga >> 32) & 0x1FFFFFFull) | (2u << 30);

    // ---- D# group 1 (256b): data_size=4B, tensor 1024x1024, tile nc x nr,
    //      dim0 stride 1024, dim1 stride 1024*1024 (unused for 2D tile).
    const unsigned long long s0 = 1024ull;
    const unsigned long long s1 = 1024ull * 1024ull;
    const u32 td0 = 1024u, td1 = 1024u;
    const u32 w0 = 2u << 16;                               // data_size=2 -> 4B
    const u32 w1 = (td0 & 0xFFFFu) << 16;                  // tensor_dim0[15:0]
    const u32 w2 = (td0 >> 16) | ((td1 & 0xFFFFu) << 16);  // td0 hi | td1 lo
    const u32 w3 = (td1 >> 16) | ((u32)nc << 16);          // td1 hi | tile_dim0
    const u32 w4 = (u32)nr;                                // tile_dim1 | tile_dim2=0
    const u32 w5 = (u32)(s0 & 0xFFFFFFFFull);              // dim0_stride[31:0]
    const u32 w6 = (u32)(s0 >> 32) | ((u32)(s1 & 0xFFFFull) << 16);
    const u32 w7 = (u32)(s1 >> 16);                        // dim1_stride[47:16]
    const i32x8 g1 = {(int)w0, (int)w1, (int)w2, (int)w3,
                      (int)w4, (int)w5, (int)w6, (int)w7};
    const i32x4 z4 = {0, 0, 0, 0};
#if __has_include(<hip/amd_detail/amd_gfx1250_TDM.h>)
    const i32x8 z8 = {0, 0, 0, 0, 0, 0, 0, 0};
    __builtin_amdgcn_tensor_load_to_lds(g0, g1, z4, z4, z8, 0);
#else
    __builtin_amdgcn_tensor_load_to_lds(g0, g1, z4, z4, 0);
#endif
    __builtin_amdgcn_s_wait_tensorcnt(0);
  }
  __syncthreads();

  float* po = patch + ((long long)bc << 14);
#pragma unroll
  for (int k = 0; k < 64; ++k) {
    const int p  = (k << 8) + (int)threadIdx.x;  // 0..16383
    const int i  = p >> 7;
    const int j  = p & 127;
    const int ri = i - pt;
    const int ci = j - pl;
    float v = 0.f;
    if ((u32)ri < (u32)nr && (u32)ci < (u32)nc)
      v = tile[ri * nc + ci];
    po[p] = v;
  }
}

extern "C" void kernel_launch(void* const* d_in, const int* in_sizes, int n_in,
                              void* d_out, int out_size, void* d_ws, size_t ws_size,
                              hipStream_t stream) {
  const float* x = (const float*)d_in[0];     // (32,3,1024,1024) f32
  const int*   l = (const int*)d_in[1];       // (32,2) i32
  float* full  = (float*)d_out;               // first output, 100,663,296 f32
  float* patch = full + (size_t)FULL_ELEMS;   // second output, 1,572,864 f32

  k_full<<<dim3(FULL_ELEMS / 4096), dim3(256), 0, stream>>>(x, l, full);
  k_patch<<<dim3(96), dim3(256), 0, stream>>>(x, l, patch);
}